// TensorTrain_71648644432008
// MI455X (gfx1250) — compile-verified
//
#include <hip/hip_runtime.h>
#include <hip/hip_bf16.h>

typedef __attribute__((ext_vector_type(2))) float v2f;
typedef __attribute__((ext_vector_type(8))) float v8f;

// ---------------------------------------------------------------------------
// Transpose a TT core (32, 512, 32) laid out [a][d][b] into [(a*32+b)][d]
// so the GEMM A-operand is row-major with contiguous K (=d).
// ---------------------------------------------------------------------------
__global__ __launch_bounds__(256)
void tt_transpose_core(const float* __restrict__ in, float* __restrict__ out) {
    int idx = blockIdx.x * 256 + threadIdx.x;        // 32*512*32 = 524288 total
    int b = idx & 31;
    int d = (idx >> 5) & 511;
    int a = idx >> 14;
    out[(size_t)((a << 5) + b) * 512 + d] = in[idx]; // coalesced read, scattered write (2MB once)
}

// ---------------------------------------------------------------------------
// fp32 WMMA GEMM: C(MxN) = A(MxK) * B(KxN), all row-major.
// Workgroup = 4 waves (128 threads), tile 64(M) x 64(N), KC = 32 staged in LDS.
// Each wave computes a 16x64 strip with 4 x v8f accumulators using
// v_wmma_f32_16x16x4_f32.
//
// Both operand tiles are stored so a lane's WMMA fragment is a contiguous,
// 8B-aligned pair of floats:
//   As[row][k]  : A frag = As[16*wave+l16][kk+2*lhalf .. +1]     (v2f)
//   Bt[col][k]  : B frag = Bt[col       ][kk+2*lhalf .. +1]     (v2f)
// -> pure ds_load_b64 fragment feeds, no VGPR shuffles before v_wmma.
// Requires K % 32 == 0, N % 64 == 0 (true here); M is guarded (M=32 GEMM).
// ---------------------------------------------------------------------------
__global__ __launch_bounds__(128)
void tt_gemm_wmma_f32(const float* __restrict__ A,
                      const float* __restrict__ B,
                      float* __restrict__ C,
                      int M, int K, int N) {
    __shared__ __align__(16) float As[64][36];   // 64 rows x 32 K (pad 36: 144B rows)
    __shared__ __align__(8)  float Bt[64][34];   // 64 cols x 32 K (pad 34: 136B rows), transposed

    const int tid   = threadIdx.x;
    const int wave  = tid >> 5;
    const int lane  = tid & 31;
    const int l16   = lane & 15;
    const int lhalf = lane >> 4;                 // 0: lanes 0-15, 1: lanes 16-31
    const int m0    = blockIdx.y * 64;
    const int n0    = blockIdx.x * 64;

    v8f acc0 = {}, acc1 = {}, acc2 = {}, acc3 = {};

    // Cooperative-fill decomposition (each thread: 16 floats = 4 x float4 global reads)
    const int aRow = (tid * 16) >> 5;            // row in 64x32 A chunk (constant per thread)
    const int aKof = (tid * 16) & 31;
    const int bRow = (tid * 16) >> 6;            // k-row in 32x64 B chunk
    const int bCof = (tid * 16) & 63;

    for (int k0 = 0; k0 < K; k0 += 32) {
        // ---- prefetch next B chunk into L2/WGP$ while this one computes
        if (k0 + 32 < K)
            __builtin_prefetch(B + (size_t)(k0 + 32 + bRow) * N + n0 + bCof, 0, 0);

        // ---- stage A chunk: rows m0..m0+63, K k0..k0+31 (rows >= M padded with zeros)
        {
            const int grow = m0 + aRow;
            #pragma unroll
            for (int i = 0; i < 16; i += 4) {
                float4 v = make_float4(0.f, 0.f, 0.f, 0.f);
                if (grow < M)
                    v = *reinterpret_cast<const float4*>(A + (size_t)grow * K + k0 + aKof + i);
                *reinterpret_cast<float4*>(&As[aRow][aKof + i]) = v;
            }
        }
        // ---- stage B chunk transposed: Bt[col][k] <- B[k0+k][n0+col]
        {
            #pragma unroll
            for (int i = 0; i < 16; i += 4) {
                float4 v = *reinterpret_cast<const float4*>(B + (size_t)(k0 + bRow) * N + n0 + bCof + i);
                Bt[bCof + i + 0][bRow] = v.x;
                Bt[bCof + i + 1][bRow] = v.y;
                Bt[bCof + i + 2][bRow] = v.z;
                Bt[bCof + i + 3][bRow] = v.w;
            }
        }
        __syncthreads();

        // ---- 8 WMMA K-steps of 4, 4 N-subtiles each; all fragments are direct v2f LDS loads
        #pragma unroll
        for (int kk = 0; kk < 32; kk += 4) {
            const int kf = kk + 2 * lhalf;
            v2f a  = *reinterpret_cast<const v2f*>(&As[16 * wave + l16][kf]);
            v2f b0 = *reinterpret_cast<const v2f*>(&Bt[ 0 + l16][kf]);
            v2f b1 = *reinterpret_cast<const v2f*>(&Bt[16 + l16][kf]);
            v2f b2 = *reinterpret_cast<const v2f*>(&Bt[32 + l16][kf]);
            v2f b3 = *reinterpret_cast<const v2f*>(&Bt[48 + l16][kf]);

            acc0 = __builtin_amdgcn_wmma_f32_16x16x4_f32(false, a, false, b0, (short)0, acc0, false, false);
            acc1 = __builtin_amdgcn_wmma_f32_16x16x4_f32(false, a, false, b1, (short)0, acc1, false, false);
            acc2 = __builtin_amdgcn_wmma_f32_16x16x4_f32(false, a, false, b2, (short)0, acc2, false, false);
            acc3 = __builtin_amdgcn_wmma_f32_16x16x4_f32(false, a, false, b3, (short)0, acc3, false, false);
        }
        __syncthreads();
    }

    // ---- store: VGPR r of C/D holds M = r (lanes 0-15) / M = r+8 (lanes 16-31)
    const int rbase = m0 + 16 * wave + 8 * lhalf;
    #pragma unroll
    for (int r = 0; r < 8; ++r) {
        const int row = rbase + r;
        if (row < M) {
            float* c = C + (size_t)row * N + n0 + l16;
            c[ 0] = acc0[r];
            c[16] = acc1[r];
            c[32] = acc2[r];
            c[48] = acc3[r];
        }
    }
}

// ---------------------------------------------------------------------------
// Batched per-sample mat-vec: Vout[a, n] = sum_b Mbuf[(a*32+b), n] * Vin[b, n]
// One thread per sample n; all global reads coalesced across lanes (n-major).
// ---------------------------------------------------------------------------
__global__ __launch_bounds__(256)
void tt_chain_matvec(const float* __restrict__ Mbuf,   // (1024, N)
                     const float* __restrict__ Vin,    // (32, N)
                     float* __restrict__ Vout,         // (32, N)
                     int N) {
    const int n = blockIdx.x * 256 + threadIdx.x;
    if (n >= N) return;

    float vin[32];
    #pragma unroll
    for (int b = 0; b < 32; ++b) vin[b] = Vin[(size_t)b * N + n];

    for (int a = 0; a < 32; ++a) {
        const float* mrow = Mbuf + (size_t)(a * 32) * N + n;
        float acc = 0.f;
        #pragma unroll
        for (int b = 0; b < 32; ++b)
            acc = fmaf(mrow[(size_t)b * N], vin[b], acc);
        Vout[(size_t)a * N + n] = acc;
    }
}

// ---------------------------------------------------------------------------
// Launch: TT forward  Z(512x4096) = core0r @ ((core1r@X0) x ((core2r@X1) x (core3@X2)))
// ---------------------------------------------------------------------------
extern "C" void kernel_launch(void* const* d_in, const int* in_sizes, int n_in,
                              void* d_out, int out_size, void* d_ws, size_t ws_size,
                              hipStream_t stream) {
    const float* X0    = (const float*)d_in[0];   // (512, 4096)
    const float* X1    = (const float*)d_in[1];   // (512, 4096)
    const float* X2    = (const float*)d_in[2];   // (512, 4096)
    const float* core0 = (const float*)d_in[3];   // (1, 512, 32)  -> row-major (512 x 32)
    const float* core1 = (const float*)d_in[4];   // (32, 512, 32)
    const float* core2 = (const float*)d_in[5];   // (32, 512, 32)
    const float* core3 = (const float*)d_in[6];   // (32, 512, 1)  -> row-major (32 x 512)
    float* Z = (float*)d_out;                     // (512, 4096) row-major ('dn')

    constexpr int B  = 4096;
    constexpr int K  = 512;
    constexpr int R  = 32;
    constexpr int RR = 1024;   // 32*32
    constexpr int D  = 512;

    char* ws = (char*)d_ws;
    float* Mbuf = (float*)(ws);                                   // 1024*4096 fp32 = 16 MB (reused)
    float* T    = (float*)(ws + (size_t)RR * B * 4);              // 1024*512  fp32 =  2 MB (reused)
    float* V2   = (float*)(ws + (size_t)RR * B * 4 + (size_t)RR * K * 4);
    float* V1   = V2 + (size_t)R * B;
    float* V0   = V1 + (size_t)R * B;

    const dim3 gblk(128);
    const dim3 tgrid(2048), tblk(256);
    const dim3 vgrid(B / 256), vblk(256);

    // v2 = core3 @ X2                          (32 x 512 x 4096)
    tt_gemm_wmma_f32<<<dim3(B / 64, 1), gblk, 0, stream>>>(core3, X2, V2, R, K, B);

    // M1 = core2r @ X1                         (1024 x 512 x 4096)
    tt_transpose_core<<<tgrid, tblk, 0, stream>>>(core2, T);
    tt_gemm_wmma_f32<<<dim3(B / 64, RR / 64), gblk, 0, stream>>>(T, X1, Mbuf, RR, K, B);
    // v1[n] = M1[n] @ v2[n]
    tt_chain_matvec<<<vgrid, vblk, 0, stream>>>(Mbuf, V2, V1, B);

    // M0 = core1r @ X0  (reuse Mbuf)           (1024 x 512 x 4096)
    tt_transpose_core<<<tgrid, tblk, 0, stream>>>(core1, T);
    tt_gemm_wmma_f32<<<dim3(B / 64, RR / 64), gblk, 0, stream>>>(T, X0, Mbuf, RR, K, B);
    // v0[n] = M0[n] @ v1[n]
    tt_chain_matvec<<<vgrid, vblk, 0, stream>>>(Mbuf, V1, V0, B);

    // Z = core0 @ V0                           (512 x 32 x 4096)
    tt_gemm_wmma_f32<<<dim3(B / 64, D / 64), gblk, 0, stream>>>(core0, V0, Z, D, R, B);
}